// FastTridiagonalSolver_87033217286175
// MI455X (gfx1250) — compile-verified
//
#include <hip/hip_runtime.h>

// ---------------------------------------------------------------------------
// Problem geometry
// ---------------------------------------------------------------------------
#define NROWS 8192
#define BCOLS 4096
#define NB    (NROWS / 16)   // 512 blocks of 16 rows
#define CB    16             // 16-row blocks per chunk
#define LCH   (CB * 16)      // 256 rows per chunk
#define PCH   (NROWS / LCH)  // 32 chunks
#define COLT  (BCOLS / 16)   // 256 column tiles of width 16

// ---------------------------------------------------------------------------
// Workspace layout (float offsets). Total ~852k floats (~3.4 MB).
// ---------------------------------------------------------------------------
#define CP_OFF   0
#define AL_OFF   (NROWS)
#define BE_OFF   (2 * NROWS)
#define GA_OFF   (3 * NROWS)
#define APR_OFF  (4 * NROWS)            // chunk-relative fwd prefix prod of alpha
#define GPR_OFF  (5 * NROWS)            // chunk-relative bwd prefix prod of gamma
#define AEND_OFF (6 * NROWS)            // PCH scalars (padded)
#define GEND_OFF (6 * NROWS + 64)
#define TF_OFF   (6 * NROWS + 128)      // NB x 16x16 forward triangular ops
#define PF_OFF   (TF_OFF + NB * 256)    // NB x 16 forward carry vectors
#define TB_OFF   (PF_OFF + NB * 16)     // NB x 16x16 backward triangular ops
#define PB_OFF   (TB_OFF + NB * 256)    // NB x 16 backward carry vectors
#define CFY_OFF  (PB_OFF + NB * 16)     // y at chunk end      [PCH][BCOLS]
#define CFI_OFF  (CFY_OFF + PCH * BCOLS) // fwd carry-in       [PCH][BCOLS]
#define CBZ_OFF  (CFI_OFF + PCH * BCOLS) // z at chunk start   [PCH][BCOLS]
#define CBI_OFF  (CBZ_OFF + PCH * BCOLS) // bwd carry-in       [PCH][BCOLS]

typedef __attribute__((ext_vector_type(2))) float v2f;
typedef __attribute__((ext_vector_type(8))) float v8f;

__device__ inline v8f wmma_f32_4(v2f a, v2f b, v8f c) {
  // D(16x16,f32) = A(16x4,f32) * B(4x16,f32) + C
  return __builtin_amdgcn_wmma_f32_16x16x4_f32(false, a, false, b, (short)0, c,
                                               false, false);
}

// ---------------------------------------------------------------------------
// K1: coefficients. cp (c') via parallel 2x2 Mobius-matrix scan, then
// alpha/beta/gamma, chunk prefix products, and per-16-block triangular
// operators Tf/Pf (forward) and Tb/Pb (backward). One workgroup.
// ---------------------------------------------------------------------------
__global__ __launch_bounds__(256) void coeff_kernel(
    const float* __restrict__ a, const float* __restrict__ b,
    const float* __restrict__ c, float* ws) {
  __shared__ float Ml[256][4];
  __shared__ float Mp[256][4];
  const int t = threadIdx.x;
  const int i0 = t * 32;

  // local composition of 32 Mobius maps  u -> c_i / (b_i - a_{i-1} u)
  float m00 = 1.f, m01 = 0.f, m10 = 0.f, m11 = 1.f;
  for (int i = i0; i < i0 + 32; ++i) {
    const float ai = (i > 0) ? a[i - 1] : 0.0f;
    const float ci = (i < NROWS - 1) ? c[i] : 0.0f;
    const float bi = b[i];
    const float n00 = ci * m10;
    const float n01 = ci * m11;
    const float n10 = bi * m10 - ai * m00;
    const float n11 = bi * m11 - ai * m01;
    const float s = 1.0f / fmaxf(fabsf(n11), 1e-30f);
    m00 = n00 * s; m01 = n01 * s; m10 = n10 * s; m11 = n11 * s;
  }
  Ml[t][0] = m00; Ml[t][1] = m01; Ml[t][2] = m10; Ml[t][3] = m11;
  __syncthreads();

  if (t == 0) {  // serial prefix over 256 chunk matrices (cheap)
    float p00 = 1.f, p01 = 0.f, p10 = 0.f, p11 = 1.f;
    for (int k = 0; k < 256; ++k) {
      Mp[k][0] = p00; Mp[k][1] = p01; Mp[k][2] = p10; Mp[k][3] = p11;
      const float l00 = Ml[k][0], l01 = Ml[k][1], l10 = Ml[k][2], l11 = Ml[k][3];
      const float q00 = l00 * p00 + l01 * p10;
      const float q01 = l00 * p01 + l01 * p11;
      const float q10 = l10 * p00 + l11 * p10;
      const float q11 = l10 * p01 + l11 * p11;
      const float s = 1.0f / fmaxf(fabsf(q11), 1e-30f);
      p00 = q00 * s; p01 = q01 * s; p10 = q10 * s; p11 = q11 * s;
    }
  }
  __syncthreads();

  // replay rows with true incoming cp; emit per-row coefficients
  float u = Mp[t][1] / Mp[t][3];  // cp entering row i0 (row 0 map is constant)
  for (int i = i0; i < i0 + 32; ++i) {
    const float ai = (i > 0) ? a[i - 1] : 0.0f;
    const float ci = (i < NROWS - 1) ? c[i] : 0.0f;
    const float denom = b[i] - ai * u;
    const float beta = 1.0f / denom;
    u = ci * beta;                       // cp[i]
    ws[CP_OFF + i] = u;
    ws[BE_OFF + i] = beta;
    ws[AL_OFF + i] = -ai * beta;         // alpha[0] == 0
    ws[GA_OFF + i] = -u;                 // gamma[N-1] == 0 (c padded)
  }
  __syncthreads();

  // chunk-relative prefix products and per-chunk totals
  if (t < PCH) {
    float ap = 1.0f;
    for (int rr = 0; rr < LCH; ++rr) {
      const int i = t * LCH + rr;
      ap *= ws[AL_OFF + i];
      ws[APR_OFF + i] = ap;
    }
    ws[AEND_OFF + t] = ap;
    float gp = 1.0f;
    for (int rr = LCH - 1; rr >= 0; --rr) {
      const int i = t * LCH + rr;
      gp *= ws[GA_OFF + i];
      ws[GPR_OFF + i] = gp;
    }
    ws[GEND_OFF + t] = gp;
  }

  // per-16-block triangular operators (2 blocks per thread)
  for (int rb = t; rb < NB; rb += 256) {
    const int s = rb * 16;
    float* Tf = ws + TF_OFF + rb * 256;
    float* Tb = ws + TB_OFF + rb * 256;
    float* Pf = ws + PF_OFF + rb * 16;
    float* Pb = ws + PB_OFF + rb * 16;
    for (int e = 0; e < 256; ++e) { Tf[e] = 0.0f; Tb[e] = 0.0f; }
    // Tf[i][j] = beta[s+j] * prod_{t=j+1..i} alpha[s+t]   (j <= i)
    for (int j = 0; j < 16; ++j) {
      float w = ws[BE_OFF + s + j];
      Tf[j * 16 + j] = w;
      for (int i2 = j + 1; i2 < 16; ++i2) {
        w *= ws[AL_OFF + s + i2];
        Tf[i2 * 16 + j] = w;
      }
    }
    // Pf[i] = prod_{t=s..s+i} alpha[t]
    {
      float pf = 1.0f;
      for (int i2 = 0; i2 < 16; ++i2) { pf *= ws[AL_OFF + s + i2]; Pf[i2] = pf; }
    }
    // Tb[i][j] = prod_{t=i..j-1} gamma[s+t]   (j >= i, empty product = 1)
    for (int i2 = 0; i2 < 16; ++i2) {
      float w = 1.0f;
      Tb[i2 * 16 + i2] = 1.0f;
      for (int j = i2 + 1; j < 16; ++j) {
        w *= ws[GA_OFF + s + j - 1];
        Tb[i2 * 16 + j] = w;
      }
    }
    // Pb[i] = prod_{t=s+i..s+15} gamma[t]
    {
      float pb = 1.0f;
      for (int i2 = 15; i2 >= 0; --i2) { pb *= ws[GA_OFF + s + i2]; Pb[i2] = pb; }
    }
  }
}

// ---------------------------------------------------------------------------
// K2: forward chunk-local pass. One wave per (chunk p, 16-col tile q).
// For each 16-row block: Y = Tf * Dblk (WMMA) + Pf * carry; carry via shfl of
// the block's row-15. Writes y (local dp) into d_out, chunk-end y to CFY.
// ---------------------------------------------------------------------------
__global__ __launch_bounds__(256) void fwd_local_kernel(
    const float* __restrict__ d, const float* __restrict__ ws,
    float* __restrict__ y, float* __restrict__ yend) {
  const int lane = threadIdx.x & 31;
  const int hi = lane >> 4;      // half-wave
  const int n = lane & 15;       // column within tile (also M for A-frag)
  const int wv = blockIdx.x * 8 + (threadIdx.x >> 5);
  const int p = wv >> 8;         // chunk  (wv / COLT)
  const int q = wv & 255;        // col tile
  const int col = (q << 4) + n;
  const float* Tf = ws + TF_OFF;
  const float* Pf = ws + PF_OFF;
  float ylast = 0.0f;

  for (int blk = 0; blk < CB; ++blk) {
    const int r = p * CB + blk;
    const int row0 = r << 4;
    v8f acc = {};
#pragma unroll
    for (int kb = 0; kb < 4; ++kb) {
      const int k0 = (kb << 2) + (hi << 1);
      // A frag: lanes0-15 M=n K={4kb,4kb+1}; lanes16-31 M=n K={4kb+2,4kb+3}
      const v2f afrag = *(const v2f*)(Tf + r * 256 + n * 16 + k0);
      // B frag: VGPR v, lanes0-15 K=4kb+v; lanes16-31 K=4kb+2+v; N=n
      v2f bfrag;
      bfrag.x = d[(row0 + k0) * BCOLS + col];
      bfrag.y = d[(row0 + k0 + 1) * BCOLS + col];
      if (blk + 1 < CB)
        __builtin_prefetch(d + (row0 + 16 + k0) * BCOLS + col, 0, 1);
      acc = wmma_f32_4(afrag, bfrag, acc);
    }
    // add carry contribution Pf[m] * ylast  (C layout: M = v + 8*hi, N = n)
    const float4 pf0 = *(const float4*)(Pf + r * 16 + hi * 8);
    const float4 pf1 = *(const float4*)(Pf + r * 16 + hi * 8 + 4);
    acc[0] += pf0.x * ylast; acc[1] += pf0.y * ylast;
    acc[2] += pf0.z * ylast; acc[3] += pf0.w * ylast;
    acc[4] += pf1.x * ylast; acc[5] += pf1.y * ylast;
    acc[6] += pf1.z * ylast; acc[7] += pf1.w * ylast;
    // new carry = row 15 of this block = acc[7] held in lanes 16..31
    ylast = __shfl(acc[7], 16 + n, 32);
#pragma unroll
    for (int v = 0; v < 8; ++v)
      y[(row0 + hi * 8 + v) * BCOLS + col] = acc[v];
  }
  if (hi == 0) yend[p * BCOLS + col] = ylast;
}

// ---------------------------------------------------------------------------
// K3: forward carry scan across PCH=32 chunks, per column.
// ---------------------------------------------------------------------------
__global__ __launch_bounds__(256) void scan_fwd_kernel(
    const float* __restrict__ yend, const float* __restrict__ ws,
    float* __restrict__ cfi) {
  const int col = blockIdx.x * 256 + threadIdx.x;
  float carry = 0.0f;
  cfi[col] = 0.0f;
  for (int p = 1; p < PCH; ++p) {
    carry = fmaf(ws[AEND_OFF + p - 1], carry, yend[(p - 1) * BCOLS + col]);
    cfi[p * BCOLS + col] = carry;
  }
}

// ---------------------------------------------------------------------------
// K4: fix dp = y + APref*carryIn on load, then backward chunk-local pass
// (blocks in reverse): Z = Tb * DPblk (WMMA) + Pb * zcarry. In-place in d_out.
// Writes chunk-start z to CBZ.
// ---------------------------------------------------------------------------
__global__ __launch_bounds__(256) void bwd_local_kernel(
    float* yz, const float* __restrict__ ws, const float* __restrict__ cfi,
    float* __restrict__ zstart) {
  const int lane = threadIdx.x & 31;
  const int hi = lane >> 4;
  const int n = lane & 15;
  const int wv = blockIdx.x * 8 + (threadIdx.x >> 5);
  const int p = wv >> 8;
  const int q = wv & 255;
  const int col = (q << 4) + n;
  const float* Tb = ws + TB_OFF;
  const float* Pb = ws + PB_OFF;
  const float* APr = ws + APR_OFF;
  const float cf = cfi[p * BCOLS + col];
  float zc = 0.0f;

  for (int blk = CB - 1; blk >= 0; --blk) {
    const int r = p * CB + blk;
    const int row0 = r << 4;
    v8f acc = {};
#pragma unroll
    for (int kb = 0; kb < 4; ++kb) {
      const int k0 = (kb << 2) + (hi << 1);
      const v2f afrag = *(const v2f*)(Tb + r * 256 + n * 16 + k0);
      const float y0 = yz[(row0 + k0) * BCOLS + col];
      const float y1 = yz[(row0 + k0 + 1) * BCOLS + col];
      v2f bfrag;  // dp = y + APref * forward carry
      bfrag.x = fmaf(APr[row0 + k0], cf, y0);
      bfrag.y = fmaf(APr[row0 + k0 + 1], cf, y1);
      if (blk > 0)
        __builtin_prefetch(yz + (row0 - 16 + k0) * BCOLS + col, 0, 1);
      acc = wmma_f32_4(afrag, bfrag, acc);
    }
    const float4 pb0 = *(const float4*)(Pb + r * 16 + hi * 8);
    const float4 pb1 = *(const float4*)(Pb + r * 16 + hi * 8 + 4);
    acc[0] += pb0.x * zc; acc[1] += pb0.y * zc;
    acc[2] += pb0.z * zc; acc[3] += pb0.w * zc;
    acc[4] += pb1.x * zc; acc[5] += pb1.y * zc;
    acc[6] += pb1.z * zc; acc[7] += pb1.w * zc;
    // new carry = row 0 of this block = acc[0] held in lanes 0..15
    zc = __shfl(acc[0], n, 32);
#pragma unroll
    for (int v = 0; v < 8; ++v)
      yz[(row0 + hi * 8 + v) * BCOLS + col] = acc[v];
  }
  if (hi == 0) zstart[p * BCOLS + col] = zc;
}

// ---------------------------------------------------------------------------
// K5: backward carry scan (reverse over chunks), per column.
// ---------------------------------------------------------------------------
__global__ __launch_bounds__(256) void scan_bwd_kernel(
    const float* __restrict__ zstart, const float* __restrict__ ws,
    float* __restrict__ cbi) {
  const int col = blockIdx.x * 256 + threadIdx.x;
  float carry = 0.0f;
  cbi[(PCH - 1) * BCOLS + col] = 0.0f;
  for (int p = PCH - 2; p >= 0; --p) {
    carry = fmaf(ws[GEND_OFF + p + 1], carry, zstart[(p + 1) * BCOLS + col]);
    cbi[p * BCOLS + col] = carry;
  }
}

// ---------------------------------------------------------------------------
// K6: final elementwise fix  x = z + GPref[row] * backwardCarry[chunk][col]
// ---------------------------------------------------------------------------
__global__ __launch_bounds__(256) void final_fix_kernel(
    float* __restrict__ x, const float* __restrict__ ws,
    const float* __restrict__ cbi) {
  const int idx4 = blockIdx.x * 256 + threadIdx.x;  // one float4 each
  const int row = idx4 >> 10;                       // / (BCOLS/4)
  const int col = (idx4 & 1023) << 2;
  const int p = row >> 8;                           // / LCH
  const float g = ws[GPR_OFF + row];
  float4 z = *(const float4*)(x + row * BCOLS + col);
  const float4 xc = *(const float4*)(cbi + p * BCOLS + col);
  z.x = fmaf(g, xc.x, z.x);
  z.y = fmaf(g, xc.y, z.y);
  z.z = fmaf(g, xc.z, z.z);
  z.w = fmaf(g, xc.w, z.w);
  *(float4*)(x + row * BCOLS + col) = z;
}

// ---------------------------------------------------------------------------
// Host entry
// ---------------------------------------------------------------------------
extern "C" void kernel_launch(void* const* d_in, const int* in_sizes, int n_in,
                              void* d_out, int out_size, void* d_ws,
                              size_t ws_size, hipStream_t stream) {
  (void)in_sizes; (void)n_in; (void)out_size; (void)ws_size;
  const float* a = (const float*)d_in[0];
  const float* b = (const float*)d_in[1];
  const float* c = (const float*)d_in[2];
  const float* d = (const float*)d_in[3];
  float* x = (float*)d_out;
  float* ws = (float*)d_ws;

  coeff_kernel<<<1, 256, 0, stream>>>(a, b, c, ws);
  fwd_local_kernel<<<(PCH * COLT) / 8, 256, 0, stream>>>(d, ws, x,
                                                         ws + CFY_OFF);
  scan_fwd_kernel<<<BCOLS / 256, 256, 0, stream>>>(ws + CFY_OFF, ws,
                                                   ws + CFI_OFF);
  bwd_local_kernel<<<(PCH * COLT) / 8, 256, 0, stream>>>(x, ws, ws + CFI_OFF,
                                                         ws + CBZ_OFF);
  scan_bwd_kernel<<<BCOLS / 256, 256, 0, stream>>>(ws + CBZ_OFF, ws,
                                                   ws + CBI_OFF);
  final_fix_kernel<<<(NROWS * BCOLS / 4) / 256, 256, 0, stream>>>(x, ws,
                                                                  ws + CBI_OFF);
}